// VoxelBackBone8x_43731357007909
// MI455X (gfx1250) — compile-verified
//
#include <hip/hip_runtime.h>

typedef __attribute__((ext_vector_type(2))) float v2f;
typedef __attribute__((ext_vector_type(8))) float v8f;

// ---------------------------------------------------------------- utilities

__global__ void fill_zero_kernel(float* __restrict__ p, long n) {
  long i = (long)blockIdx.x * blockDim.x + threadIdx.x;
  if (i < n) p[i] = 0.0f;
}

// f[n, c] = sum_p voxels[n, p, c] / max(num_points[n], 1)
__global__ void voxel_mean_kernel(const float* __restrict__ vox,
                                  const int* __restrict__ npnt,
                                  float* __restrict__ f, int n0) {
  int i = blockIdx.x * blockDim.x + threadIdx.x;
  if (i >= n0 * 4) return;
  int n = i >> 2, c = i & 3;
  float s = 0.0f;
#pragma unroll
  for (int p = 0; p < 5; ++p) s += vox[(n * 5 + p) * 4 + c];
  float d = (float)npnt[n];
  if (d < 1.0f) d = 1.0f;
  f[i] = s / d;
}

// y = relu(x * s[c] + b[c]), in place, rows [0, n)
__global__ void bnrelu_kernel(float* __restrict__ x, const float* __restrict__ s,
                              const float* __restrict__ b, long n, int C) {
  long i = (long)blockIdx.x * blockDim.x + threadIdx.x;
  if (i >= n * C) return;
  int c = (int)(i % C);
  float v = x[i] * s[c] + b[c];
  x[i] = v > 0.0f ? v : 0.0f;
}

// segment max over batch ids; valid because inputs are post-ReLU (>= 0) and
// the output buffer is zero-initialized: uint compare == float compare here.
__global__ void segmax_kernel(const float* __restrict__ x, const int* __restrict__ batch,
                              float* __restrict__ out, long n, int C) {
  long i = (long)blockIdx.x * blockDim.x + threadIdx.x;
  if (i >= n * C) return;
  long r = i / C;
  int c = (int)(i % C);
  unsigned bits = __float_as_uint(x[i]);
  atomicMax(reinterpret_cast<unsigned int*>(out) + (long)batch[r] * C + c, bits);
}

// ------------------------------------------------- sparse conv via f32 WMMA
//
// One wave32 per (16-row tile, tap k). The wave gathers its A fragments
// (16 x CIN f32) ONCE into registers, preloads the 16 scatter rows, then
// loops over the COUT/16 column tiles reusing A from registers — this is the
// gather-bandwidth-optimal schedule for a gather-bound, L2-resident workload.
//
//   A (16x4, f32): lanes 0-15 -> M=lane, K=0(.x),1(.y); lanes 16-31 -> K=2,3
//   B (4x16, f32): lane half selects K-pair {0,1} / {2,3}; N = lane & 15
//   C/D (16x16): VGPR i -> M = i + 8*(lane>=16), N = lane & 15
template <int CIN, int COUT>
__global__ __launch_bounds__(32) void spconv_wmma_kernel(
    const float* __restrict__ feat,   // (nin, CIN)
    const float* __restrict__ W,      // (K, CIN, COUT)
    const int* __restrict__ rin,      // (K, R)
    const int* __restrict__ rout,     // (K, R)
    float* __restrict__ out,          // (nout+1, COUT), zero-initialized
    int R) {
  const int tile = blockIdx.x;
  const int k    = blockIdx.y;
  const int lane = threadIdx.x & 31;
  const int m    = lane & 15;
  const int hi   = lane >> 4;  // which lane-half (selects K pair / M+8)

  const long rbase = (long)k * R + (long)tile * 16;
  const int  rA    = tile * 16 + m;
  const int  rowA  = (rA < R) ? rin[rbase + m] : 0;  // padded entries -> row 0

  const float* __restrict__ wk = W + (long)k * CIN * COUT;
  const float* __restrict__ fr = feat + (long)rowA * CIN;

  // Gather A fragments once (register-resident across all column tiles).
  v2f a[CIN / 4];
#pragma unroll
  for (int kk = 0; kk < CIN / 4; ++kk) {
    const int kb = kk * 4 + 2 * hi;
    a[kk].x = fr[kb];
    a[kk].y = fr[kb + 1];
  }

  // Preload scatter rows (sentinel 'nout' rulebook entries land in dump row;
  // -1 marks past-the-end tile padding and is skipped).
  int orow[8];
#pragma unroll
  for (int i = 0; i < 8; ++i) {
    const int M = i + 8 * hi;
    const int r = tile * 16 + M;
    orow[i] = (r < R) ? rout[rbase + M] : -1;
  }

  for (int ct = 0; ct < COUT / 16; ++ct) {
    const int n = ct * 16 + m;
    v8f acc = {};
#pragma unroll
    for (int kk = 0; kk < CIN / 4; ++kk) {
      const int kb = kk * 4 + 2 * hi;
      v2f b;
      b.x = wk[(long)kb * COUT + n];
      b.y = wk[(long)(kb + 1) * COUT + n];
      acc = __builtin_amdgcn_wmma_f32_16x16x4_f32(
          /*neg_a=*/false, a[kk], /*neg_b=*/false, b,
          /*c_mod=*/(short)0, acc, /*reuse_a=*/false, /*reuse_b=*/false);
    }
#pragma unroll
    for (int i = 0; i < 8; ++i) {
      if (orow[i] >= 0) {
        atomicAdd(&out[(long)orow[i] * COUT + n], acc[i]);
      }
    }
  }
}

// ---------------------------------------------------------------- launcher

static inline long cdiv(long a, long b) { return (a + b - 1) / b; }

#define CONV_LAYER(CIN, COUT, KT, fin, rinp, routp, Rr, nout, wp, sp, bp, fout)      \
  do {                                                                               \
    long tot = (long)((nout) + 1) * (COUT);                                          \
    fill_zero_kernel<<<cdiv(tot, 256), 256, 0, stream>>>((fout), tot);               \
    dim3 g((unsigned)cdiv((Rr), 16), (unsigned)(KT), 1u);                            \
    spconv_wmma_kernel<CIN, COUT>                                                    \
        <<<g, 32, 0, stream>>>((fin), (wp), (rinp), (routp), (fout), (Rr));          \
    long tb = (long)(nout) * (COUT);                                                 \
    bnrelu_kernel<<<cdiv(tb, 256), 256, 0, stream>>>((fout), (sp), (bp),             \
                                                     (long)(nout), (COUT));         \
  } while (0)

extern "C" void kernel_launch(void* const* d_in, const int* in_sizes, int n_in,
                              void* d_out, int out_size, void* d_ws, size_t ws_size,
                              hipStream_t stream) {
  // ---- inputs (setup_inputs dict order) ----
  const float* voxels = (const float*)d_in[0];
  const int*   vnp    = (const int*)d_in[1];
  const int* rb1_in  = (const int*)d_in[3];
  const int* rb1_out = (const int*)d_in[4];
  const int* rb2s_in = (const int*)d_in[5];
  const int* rb2s_out= (const int*)d_in[6];
  const int* rb2_in  = (const int*)d_in[8];
  const int* rb2_out = (const int*)d_in[9];
  const int* rb3s_in = (const int*)d_in[10];
  const int* rb3s_out= (const int*)d_in[11];
  const int* rb3_in  = (const int*)d_in[13];
  const int* rb3_out = (const int*)d_in[14];
  const int* rb4s_in = (const int*)d_in[15];
  const int* rb4s_out= (const int*)d_in[16];
  const int* rb4_in  = (const int*)d_in[18];
  const int* rb4_out = (const int*)d_in[19];
  const int* rbo_in  = (const int*)d_in[20];
  const int* rbo_out = (const int*)d_in[21];
  const int* batch_out = (const int*)d_in[22];

  const int N0 = in_sizes[1];
  const int R1  = in_sizes[3]  / 27;
  const int R2s = in_sizes[5]  / 27;
  const int n2  = in_sizes[7];
  const int R2  = in_sizes[8]  / 27;
  const int R3s = in_sizes[10] / 27;
  const int n3  = in_sizes[12];
  const int R3  = in_sizes[13] / 27;
  const int R4s = in_sizes[15] / 27;
  const int n4  = in_sizes[17];
  const int R4  = in_sizes[18] / 27;
  const int Ro  = in_sizes[20] / 3;
  const int no  = in_sizes[22];

  const float* w_in = (const float*)d_in[23];
  const float* s_in = (const float*)d_in[24];
  const float* b_in = (const float*)d_in[25];
  const float* w1   = (const float*)d_in[26];
  const float* s1   = (const float*)d_in[27];
  const float* b1   = (const float*)d_in[28];
  const float* w2a  = (const float*)d_in[29];
  const float* s2a  = (const float*)d_in[30];
  const float* b2a  = (const float*)d_in[31];
  const float* w2b  = (const float*)d_in[32];
  const float* s2b  = (const float*)d_in[33];
  const float* b2b  = (const float*)d_in[34];
  const float* w2c  = (const float*)d_in[35];
  const float* s2c  = (const float*)d_in[36];
  const float* b2c  = (const float*)d_in[37];
  const float* w3a  = (const float*)d_in[38];
  const float* s3a  = (const float*)d_in[39];
  const float* b3a  = (const float*)d_in[40];
  const float* w3b  = (const float*)d_in[41];
  const float* s3b  = (const float*)d_in[42];
  const float* b3b  = (const float*)d_in[43];
  const float* w3c  = (const float*)d_in[44];
  const float* s3c  = (const float*)d_in[45];
  const float* b3c  = (const float*)d_in[46];
  const float* w4a  = (const float*)d_in[47];
  const float* s4a  = (const float*)d_in[48];
  const float* b4a  = (const float*)d_in[49];
  const float* w4b  = (const float*)d_in[50];
  const float* s4b  = (const float*)d_in[51];
  const float* b4b  = (const float*)d_in[52];
  const float* w4c  = (const float*)d_in[53];
  const float* s4c  = (const float*)d_in[54];
  const float* b4c  = (const float*)d_in[55];
  const float* w_out = (const float*)d_in[56];
  const float* s_out = (const float*)d_in[57];
  const float* b_out = (const float*)d_in[58];

  // ---- workspace ping-pong buffers ----
  size_t E = (size_t)N0 * 4;
  auto mx = [](size_t a, size_t b) { return a > b ? a : b; };
  E = mx(E, (size_t)(N0 + 1) * 16);
  E = mx(E, (size_t)(n2 + 1) * 32);
  E = mx(E, (size_t)(n3 + 1) * 64);
  E = mx(E, (size_t)(n4 + 1) * 64);
  E = mx(E, (size_t)(no + 1) * 128);
  float* bufA = (float*)d_ws;
  float* bufB = bufA + E;
  (void)ws_size; (void)n_in;

  // ---- stage 0: voxel feature mean -> bufA (N0 x 4) ----
  voxel_mean_kernel<<<cdiv((long)N0 * 4, 256), 256, 0, stream>>>(voxels, vnp, bufA, N0);

  // ---- sparse conv stack (gather - f32 WMMA - atomic scatter) ----
  CONV_LAYER( 4,  16, 27, bufA, rb1_in,  rb1_out,  R1,  N0, w_in, s_in, b_in, bufB);
  CONV_LAYER(16,  16, 27, bufB, rb1_in,  rb1_out,  R1,  N0, w1,   s1,   b1,   bufA);
  CONV_LAYER(16,  32, 27, bufA, rb2s_in, rb2s_out, R2s, n2, w2a,  s2a,  b2a,  bufB);
  CONV_LAYER(32,  32, 27, bufB, rb2_in,  rb2_out,  R2,  n2, w2b,  s2b,  b2b,  bufA);
  CONV_LAYER(32,  32, 27, bufA, rb2_in,  rb2_out,  R2,  n2, w2c,  s2c,  b2c,  bufB);
  CONV_LAYER(32,  64, 27, bufB, rb3s_in, rb3s_out, R3s, n3, w3a,  s3a,  b3a,  bufA);
  CONV_LAYER(64,  64, 27, bufA, rb3_in,  rb3_out,  R3,  n3, w3b,  s3b,  b3b,  bufB);
  CONV_LAYER(64,  64, 27, bufB, rb3_in,  rb3_out,  R3,  n3, w3c,  s3c,  b3c,  bufA);
  CONV_LAYER(64,  64, 27, bufA, rb4s_in, rb4s_out, R4s, n4, w4a,  s4a,  b4a,  bufB);
  CONV_LAYER(64,  64, 27, bufB, rb4_in,  rb4_out,  R4,  n4, w4b,  s4b,  b4b,  bufA);
  CONV_LAYER(64,  64, 27, bufA, rb4_in,  rb4_out,  R4,  n4, w4c,  s4c,  b4c,  bufB);
  CONV_LAYER(64, 128,  3, bufB, rbo_in,  rbo_out,  Ro,  no, w_out, s_out, b_out, bufA);

  // ---- segment max over batch -> d_out (B x 128) ----
  fill_zero_kernel<<<cdiv((long)out_size, 256), 256, 0, stream>>>((float*)d_out, (long)out_size);
  segmax_kernel<<<cdiv((long)no * 128, 256), 256, 0, stream>>>(bufA, batch_out,
                                                               (float*)d_out, (long)no, 128);
}